// BoundaryLoss_3659312136379
// MI455X (gfx1250) — compile-verified
//
#include <hip/hip_runtime.h>
#include <math.h>
#include <stdint.h>

// BoundaryLoss for pred[8,1,256,256] f32, target[8,1,256,256] i32 -> scalar f32.
// Pass A: vertical 1D EDT scans (store d1^2). Pass B: horizontal min-envelope
// + fused sigmoid/err^2 + per-row reduction. Pass C: final reduction + mean.
// CDNA5 paths: V_WMMA_F32_16X16X4_F32 reductions + GLOBAL_LOAD_ASYNC_TO_LDS_B128
// row staging (guarded; falls back to sync staging if builtins absent).

typedef float v2f __attribute__((ext_vector_type(2)));
typedef float v8f __attribute__((ext_vector_type(8)));
typedef int   v4i __attribute__((vector_size(16)));   // matches builtin pointee

#if defined(__has_builtin)
# if __has_builtin(__builtin_amdgcn_global_load_async_to_lds_b128) && \
     __has_builtin(__builtin_amdgcn_s_wait_asynccnt)
#  define HAVE_ASYNC_LDS 1
# endif
#endif

#define BATCH 8
#define HDIM 256
#define WDIM 256
#define NPIX (BATCH * HDIM * WDIM)   // 524288
#define NROWS (BATCH * HDIM)         // 2048
#define BIGF 1.0e6f

// ---------------- Pass A: per-column vertical distance (stores d1^2) --------
__global__ __launch_bounds__(256) void edt_vertical_kernel(
    const int* __restrict__ target, float* __restrict__ d1sq)
{
    const int b = blockIdx.x;        // image
    const int w = threadIdx.x;       // column
    const int base = b * HDIM * WDIM + w;

    // forward scan: distance to most recent zero above (or BIG)
    int last = -1;
    for (int h = 0; h < HDIM; ++h) {
        int t = target[base + h * WDIM];
        if (t == 0) last = h;
        float df = (last >= 0) ? (float)(h - last) : BIGF;
        d1sq[base + h * WDIM] = df;
    }
    // backward scan: min with distance to next zero below, then square
    int nxt = -1;
    for (int h = HDIM - 1; h >= 0; --h) {
        int t = target[base + h * WDIM];
        if (t == 0) nxt = h;
        float db = (nxt >= 0) ? (float)(nxt - h) : BIGF;
        float df = d1sq[base + h * WDIM];
        float d  = fminf(df, db);
        d1sq[base + h * WDIM] = d * d;   // same rounding as reference d1**2
    }
}

// ---------------- WMMA-based deterministic block sum (blockDim == 256) ------
// Each wave: lane L contributes v into A[L%16][2*(L>=16)] of a 16x4 f32 A;
// B = all ones -> D[m][n] = v_m + v_{m+16}. Per-lane sum of the 8 C regs gives
// rows 0-7 (lanes 0-15) / rows 8-15 (lanes 16-31); lane0+lane16 = wave total.
__device__ __forceinline__ float block_sum_wmma(float v, float* sW)
{
    const int tid  = threadIdx.x;
    const int lane = tid & 31;
    const int wid  = tid >> 5;       // 8 waves per block

    v2f a; a[0] = v;    a[1] = 0.0f;
    v2f o; o[0] = 1.0f; o[1] = 1.0f;
    v8f c = {};
    c = __builtin_amdgcn_wmma_f32_16x16x4_f32(
            /*neg_a=*/false, a, /*neg_b=*/false, o,
            /*c_mod=*/(short)0, c, /*reuse_a=*/false, /*reuse_b=*/false);
    float t = c[0] + c[1] + c[2] + c[3] + c[4] + c[5] + c[6] + c[7];

    if (lane == 0)  sW[2 * wid]     = t;   // rows 0-7 of this wave
    if (lane == 16) sW[2 * wid + 1] = t;   // rows 8-15 of this wave
    __syncthreads();

    float total = 0.0f;
    if (tid == 0) {
        #pragma unroll
        for (int i = 0; i < 16; ++i) total += sW[i];  // fixed order: deterministic
    }
    return total;   // valid on tid==0 only
}

// ---------------- Pass B: horizontal lower envelope + fused loss ------------
__global__ __launch_bounds__(256) void row_loss_kernel(
    const float* __restrict__ pred, const int* __restrict__ target,
    const float* __restrict__ d1sq, float* __restrict__ partials)
{
    __shared__ float sD[WDIM];
    __shared__ float sW[16];

    const int row = blockIdx.x;          // b*H + h
    const int j   = threadIdx.x;         // output column
    const int idx = row * WDIM + j;

    float p  = pred[idx];
    float tf = (float)target[idx];

#ifdef HAVE_ASYNC_LDS
    // Stage the 1KB d1^2 row into LDS with the async (memory->LDS) path:
    // 64 lanes x b128, tracked by ASYNCcnt.
    if (j < WDIM / 4) {
        typedef __attribute__((address_space(1))) v4i GV4;
        typedef __attribute__((address_space(3))) v4i LV4;
        GV4* g = (GV4*)(uintptr_t)(const void*)&d1sq[row * WDIM + 4 * j];
        LV4* l = (LV4*)(uint32_t)(uintptr_t)(void*)&sD[4 * j];
        __builtin_amdgcn_global_load_async_to_lds_b128(g, l, 0, 0);
    }
    __builtin_amdgcn_s_wait_asynccnt(0);
#else
    sD[j] = d1sq[idx];
#endif
    __syncthreads();

    const float jf = (float)j;
    // 4 independent min-accumulators (exact: min is associative) for ILP;
    // float4 LDS reads -> ds_load_b128 broadcasts.
    float b0 = 3.0e38f, b1 = 3.0e38f, b2 = 3.0e38f, b3 = 3.0e38f;
    const float4* sD4 = (const float4*)sD;
    #pragma unroll 8
    for (int k4 = 0; k4 < WDIM / 4; ++k4) {
        float4 s4 = sD4[k4];
        float d0 = jf - (float)(4 * k4);   // exact integer-valued f32
        float d1 = d0 - 1.0f;
        float d2 = d0 - 2.0f;
        float d3 = d0 - 3.0f;
        b0 = fminf(b0, fmaf(d0, d0, s4.x));  // == (j-k)^2 + d1^2 rounding
        b1 = fminf(b1, fmaf(d1, d1, s4.y));
        b2 = fminf(b2, fmaf(d2, d2, s4.z));
        b3 = fminf(b3, fmaf(d3, d3, s4.w));
    }
    float best = fminf(fminf(b0, b1), fminf(b2, b3));

    float gt  = sqrtf(best);
    float sig = 1.0f / (1.0f + expf(-p));
    float e   = sig - tf;

    float total = block_sum_wmma(gt * e * e, sW);
    if (j == 0) partials[row] = total;
}

// ---------------- Pass C: final deterministic reduction + mean --------------
__global__ __launch_bounds__(256) void final_reduce_kernel(
    const float* __restrict__ partials, float* __restrict__ out)
{
    __shared__ float sW[16];
    const int tid = threadIdx.x;

    float v = 0.0f;
    #pragma unroll
    for (int i = 0; i < NROWS / 256; ++i)   // 8 partials per thread, fixed order
        v += partials[tid + i * 256];

    float total = block_sum_wmma(v, sW);
    if (tid == 0) out[0] = total * (1.0f / (float)NPIX);
}

// ---------------------------------------------------------------------------
extern "C" void kernel_launch(void* const* d_in, const int* in_sizes, int n_in,
                              void* d_out, int out_size, void* d_ws, size_t ws_size,
                              hipStream_t stream)
{
    const float* pred   = (const float*)d_in[0];
    const int*   target = (const int*)d_in[1];

    float* d1sq     = (float*)d_ws;        // NPIX floats  (2 MB)
    float* partials = d1sq + NPIX;         // NROWS floats (8 KB)

    edt_vertical_kernel<<<BATCH, WDIM, 0, stream>>>(target, d1sq);
    row_loss_kernel<<<NROWS, WDIM, 0, stream>>>(pred, target, d1sq, partials);
    final_reduce_kernel<<<1, 256, 0, stream>>>(partials, (float*)d_out);
}